// GCNLayer_80753975099751
// MI455X (gfx1250) — compile-verified
//
#include <hip/hip_runtime.h>
#include <hip/hip_bf16.h>

#define IN_DIM 128
#define OUT_DIM 128

typedef float v2f __attribute__((ext_vector_type(2)));
typedef float v8f __attribute__((ext_vector_type(8)));

// ---------------------------------------------------------------------------
// Stage 1: SpMM  H[rows[e], :] += vals[e] * X[cols[e], :]
// One thread handles one (edge, 4-feature-group): float4 gather from X
// (L2-resident, 25.6MB), 4x global_atomic_add_f32 scatter into H.
// ---------------------------------------------------------------------------
__global__ void gcn_spmm_atomic(const int* __restrict__ rows,
                                const int* __restrict__ cols,
                                const float* __restrict__ vals,
                                const float* __restrict__ X,
                                float* __restrict__ H,
                                int E) {
    int idx = blockIdx.x * blockDim.x + threadIdx.x;   // E * 32 work items
    int e = idx >> 5;                                  // edge id
    if (e >= E) return;
    int g = (idx & 31) << 2;                           // feature group base (0..124)

    int r = rows[e];
    int c = cols[e];
    float v = vals[e];

    const float4 x = *(const float4*)(X + (size_t)c * IN_DIM + g);
    float* hp = H + (size_t)r * IN_DIM + g;
    atomicAdd(hp + 0, v * x.x);
    atomicAdd(hp + 1, v * x.y);
    atomicAdd(hp + 2, v * x.z);
    atomicAdd(hp + 3, v * x.w);
}

// ---------------------------------------------------------------------------
// Stage 2: out = relu(dropout(H @ W + b))
// Block = 256 threads = 8 waves. W (128x128 f32, 64KB) staged in LDS once per
// block. Each wave owns a 16-row tile and computes all 128 output columns:
// 8 x v8f accumulators, K stepped by 4 using V_WMMA_F32_16X16X4_F32
// (fp32-exact, matches reference precision). A-fragment loaded once per K-step
// and reused across the 8 N-tiles.
// ---------------------------------------------------------------------------
__global__ void gcn_gemm_wmma(const float* __restrict__ H,
                              const float* __restrict__ W,
                              const float* __restrict__ bias,
                              const float* __restrict__ drop_u,
                              float* __restrict__ out,
                              int N) {
    __shared__ float Wlds[IN_DIM * OUT_DIM];   // 64 KB (WGP has 320 KB)

    // Cooperative float4 load of W into LDS.
    int t = threadIdx.x;
    for (int i = t * 4; i < IN_DIM * OUT_DIM; i += 256 * 4) {
        *(float4*)(Wlds + i) = *(const float4*)(W + i);
    }
    __syncthreads();

    int wave = t >> 5;
    int lane = t & 31;
    int rowTile = blockIdx.x * 8 + wave;       // 16-row tile index (wave-uniform)
    if (rowTile * 16 >= N) return;             // whole wave exits: EXEC all-1s for WMMA
    int row0 = rowTile * 16;

    // WMMA f32 16x16x4 fragment coordinates (ISA 7.12.2):
    int m     = lane & 15;                     // A row / B col within tile
    int kb    = (lane >> 4) * 2;               // K sub-offset: lanes 0-15 -> K{0,1}, 16-31 -> K{2,3}

    v8f acc[8] = {v8f{}, v8f{}, v8f{}, v8f{}, v8f{}, v8f{}, v8f{}, v8f{}};

    const float* hrow = H + (size_t)(row0 + m) * IN_DIM;

    for (int k = 0; k < IN_DIM; k += 4) {
        v2f a;
        a.x = hrow[k + kb + 0];
        a.y = hrow[k + kb + 1];
        const float* wk0 = Wlds + (k + kb + 0) * OUT_DIM;
        const float* wk1 = Wlds + (k + kb + 1) * OUT_DIM;
#pragma unroll
        for (int nt = 0; nt < 8; ++nt) {
            int n = nt * 16 + m;
            v2f b;
            b.x = wk0[n];
            b.y = wk1[n];
            acc[nt] = __builtin_amdgcn_wmma_f32_16x16x4_f32(
                /*neg_a=*/false, a, /*neg_b=*/false, b,
                /*c_mod=*/(short)0, acc[nt],
                /*reuse_a=*/false, /*reuse_b=*/false);
        }
    }

    // Epilogue: bias + dropout(p=0.5 -> mask in {0,2}) + ReLU, fused at store.
    // C/D layout: VGPR r -> row (r + 8*(lane>>4)), col = lane&15 within N-tile.
    int rhalf = (lane >> 4) * 8;
    int ncol  = lane & 15;
#pragma unroll
    for (int nt = 0; nt < 8; ++nt) {
        int n = nt * 16 + ncol;
        float bv = bias[n];
#pragma unroll
        for (int r = 0; r < 8; ++r) {
            int row = row0 + rhalf + r;
            float val = acc[nt][r] + bv;
            float u = drop_u[(size_t)row * OUT_DIM + n];
            val = (u >= 0.5f) ? (val * 2.0f) : 0.0f;
            out[(size_t)row * OUT_DIM + n] = fmaxf(val, 0.0f);
        }
    }
}

// ---------------------------------------------------------------------------
extern "C" void kernel_launch(void* const* d_in, const int* in_sizes, int n_in,
                              void* d_out, int out_size, void* d_ws, size_t ws_size,
                              hipStream_t stream) {
    const int*   rows   = (const int*)d_in[0];
    const int*   cols   = (const int*)d_in[1];
    const float* vals   = (const float*)d_in[2];
    const float* X      = (const float*)d_in[3];
    const float* W      = (const float*)d_in[4];
    const float* bias   = (const float*)d_in[5];
    const float* drop_u = (const float*)d_in[6];
    float*       out    = (float*)d_out;
    float*       H      = (float*)d_ws;      // [N, 128] f32 accumulator

    int E = in_sizes[0];
    int N = in_sizes[3] / IN_DIM;

    // Zero the SpMM accumulator every call (deterministic, capture-safe).
    hipMemsetAsync(H, 0, (size_t)N * IN_DIM * sizeof(float), stream);

    // Stage 1: SpMM via float atomics. E*32 threads (4 features each).
    long long work = (long long)E * 32;
    int blocks = (int)((work + 255) / 256);
    gcn_spmm_atomic<<<blocks, 256, 0, stream>>>(rows, cols, vals, X, H, E);

    // Stage 2: WMMA GEMM + fused bias/dropout/relu. 8 row-tiles per block.
    int rowTiles = (N + 15) / 16;
    int gblocks  = (rowTiles + 7) / 8;
    gcn_gemm_wmma<<<gblocks, 256, 0, stream>>>(H, W, bias, drop_u, out, N);
}